// TopKRouter_58145267253887
// MI455X (gfx1250) — compile-verified
//
#include <hip/hip_runtime.h>
#include <hip/hip_bf16.h>
#include <stdint.h>

// Top-8 MoE router for MI455X (gfx1250).
// Strategy: TDM (tensor_load_to_lds) stages a 128-row x 64-col fp32 tile of
// router logits into LDS with hardware padding (row stride 68 dwords -> 16B
// aligned rows, bank-spread). Each thread then does a register-resident top-8
// selection on its row. Softmax + renorm collapses to exp(v-max)/sum over the
// 8 selected logits, so only 8 transcendentals per row are needed.

typedef __attribute__((ext_vector_type(4))) unsigned int u32x4;
typedef __attribute__((ext_vector_type(8))) int          i32x8;
typedef __attribute__((ext_vector_type(4))) int          i32x4;

#define ROWS_PER_BLOCK 128
#define COLS           64
#define ROW_STRIDE     68   // 64 data dwords + 4 pad dwords (TDM pad_amount=4 each 64 DW)

__global__ __launch_bounds__(ROWS_PER_BLOCK) void
topk_router_kernel(const float* __restrict__ x, float* __restrict__ out, int nrows)
{
    __shared__ float smem[ROWS_PER_BLOCK * ROW_STRIDE]; // 128*68*4 = 34,816 B

    const int tid  = threadIdx.x;
    const long long base = (long long)blockIdx.x * ROWS_PER_BLOCK;
    const unsigned rem = (unsigned)((nrows - base) < ROWS_PER_BLOCK ? (nrows - base)
                                                                    : (long long)ROWS_PER_BLOCK);

    // ---- Wave 0 issues the TDM load of the whole tile, then waits TENSORcnt ----
    if (tid < 32) {
        unsigned long long gaddr = (unsigned long long)(uintptr_t)(x + base * COLS);
        // LDS byte offset of smem within the workgroup allocation (low 32 bits
        // of the flat LDS-aperture address).
        unsigned lds_off = (unsigned)((unsigned long long)(uintptr_t)(&smem[0]) & 0xFFFFFFFFull);

        u32x4 g0;
        g0[0] = 1u;                                        // count=1, no gather, user descriptor
        g0[1] = lds_off;                                   // lds_addr
        g0[2] = (unsigned)(gaddr & 0xFFFFFFFFull);         // global_addr[31:0]
        g0[3] = (unsigned)((gaddr >> 32) & 0x1FFFFFFull)   // global_addr[56:32]
              | (2u << 30);                                // type = 2 ("image")

        i32x8 g1;
        g1[0] = (int)((2u << 16)      // data_size = 4 bytes
                    | (1u << 20)      // pad_enable
                    | (5u << 22)      // pad_interval code 5 -> every 64 DWORDs
                    | (3u << 25));    // pad_amount  code 3 -> 4 DWORDs
        g1[1] = (int)((unsigned)COLS << 16);               // tensor_dim0[15:0] in [31:16]
        g1[2] = (int)(rem << 16);                          // tensor_dim0 hi=0 | tensor_dim1 lo = rem rows
        g1[3] = (int)((unsigned)COLS << 16);               // tensor_dim1 hi=0 | tile_dim0 = 64
        g1[4] = (int)(unsigned)ROWS_PER_BLOCK;             // tile_dim1 = 128 | tile_dim2 = 0
        g1[5] = COLS;                                      // tensor_dim0_stride = 64 elements
        g1[6] = 0;
        g1[7] = 0;

        i32x4 gz4 = {0, 0, 0, 0};                          // groups 2/3 unused (2D tensor)
        i32x8 gz8 = {0, 0, 0, 0, 0, 0, 0, 0};              // extra group (6-arg toolchain form)

        __builtin_amdgcn_tensor_load_to_lds(g0, g1, gz4, gz4, gz8, /*cpol=*/0);
        __builtin_amdgcn_s_wait_tensorcnt(0);
    }
    __syncthreads();

    const size_t N = (size_t)nrows;
    float* __restrict__ out_w  = out;                 // [N,8] weights
    float* __restrict__ out_id = out + N * 8;         // [N,8] ids (as float)
    float* __restrict__ out_x  = out + N * 16;        // [N,64] logits pass-through

    // ---- Per-thread: top-8 of this thread's row, register resident ----
    if ((unsigned)tid < rem) {
        const float* row = &smem[tid * ROW_STRIDE];

        float v[COLS];
        #pragma unroll
        for (int j = 0; j < COLS; ++j) v[j] = row[j];   // 16x ds_load_b128

        unsigned long long chosen = 0ull;
        float wv[8];
        int   wi[8];

        #pragma unroll
        for (int k = 0; k < 8; ++k) {
            float bv = -3.402823466e38f;
            int   bi = 0;
            #pragma unroll
            for (int j = 0; j < COLS; ++j) {
                bool taken = ((chosen >> j) & 1ull) != 0ull;
                // strict '>' => lowest index wins ties (matches jax.lax.top_k)
                if (!taken && v[j] > bv) { bv = v[j]; bi = j; }
            }
            chosen |= (1ull << bi);
            wv[k] = bv;
            wi[k] = bi;
        }

        // softmax over the selected 8 (global softmax denominator cancels)
        const float m = wv[0];
        float s = 0.0f;
        #pragma unroll
        for (int k = 0; k < 8; ++k) {
            float e = __expf(wv[k] - m);   // v_exp_f32
            wv[k] = e;
            s += e;
        }
        const float inv = 1.0f / s;

        const size_t r = (size_t)(base + tid);
        #pragma unroll
        for (int k = 0; k < 8; ++k) {
            out_w [r * 8 + k] = wv[k] * inv;
            out_id[r * 8 + k] = (float)wi[k];
        }
    }

    // ---- Coalesced pass-through copy of the logits tile ----
    const unsigned total = rem * COLS;
    for (unsigned idx = (unsigned)tid; idx < total; idx += ROWS_PER_BLOCK) {
        unsigned rr = idx >> 6;
        unsigned cc = idx & 63u;
        out_x[(size_t)base * COLS + idx] = smem[rr * ROW_STRIDE + cc];
    }
}

extern "C" void kernel_launch(void* const* d_in, const int* in_sizes, int n_in,
                              void* d_out, int out_size, void* d_ws, size_t ws_size,
                              hipStream_t stream)
{
    (void)n_in; (void)out_size; (void)d_ws; (void)ws_size;
    const float* x = (const float*)d_in[0];
    float* out = (float*)d_out;
    const int nrows = in_sizes[0] / 64;   // [N, 64] fp32 logits
    const int blocks = (nrows + ROWS_PER_BLOCK - 1) / ROWS_PER_BLOCK;
    topk_router_kernel<<<blocks, ROWS_PER_BLOCK, 0, stream>>>(x, out, nrows);
}